// RNNMotionClassifier_29892972380518
// MI455X (gfx1250) — compile-verified
//
#include <hip/hip_runtime.h>

typedef __attribute__((ext_vector_type(2))) float v2f;
typedef __attribute__((ext_vector_type(8))) float v8f;

#define SEQ   512
#define BATCH 4096
#define IN    9
#define HID   64
#define OUT   10

#define HSTR 66                 // padded LDS row stride (floats) -> conflict-free b64 frag loads
#define XSTR 12                 // x K-dim padded 9 -> 12 (3 WMMA K=4 chunks)

#define XBUFSZ (16 * XSTR)
#define HBUFSZ (16 * HSTR)
#define XOFF   0
#define H1OFF  (2 * XBUFSZ)
#define H2OFF  (H1OFF + 2 * HBUFSZ)
#define LDSFLOATS (H2OFF + 2 * HBUFSZ)   // 4608 floats = 18 KB

#if defined(__has_builtin)
#if __has_builtin(__builtin_amdgcn_sched_group_barrier)
#define SCHED_GROUP(mask, size, id) __builtin_amdgcn_sched_group_barrier((mask), (size), (id))
#endif
#endif
#ifndef SCHED_GROUP
#define SCHED_GROUP(mask, size, id)
#endif
// mask bits: 0x8 = MFMA/WMMA, 0x100 = DS read

static __device__ inline float tanh_fast(float x) {
    // tanh(x) = 1 - 2/(exp(2x)+1); exp -> v_exp_f32, rcp -> v_rcp_f32
    float e = __expf(2.0f * x);
    float r = __builtin_amdgcn_rcpf(e + 1.0f);
    return fmaf(-2.0f, r, 1.0f);
}

// A-fragment for V_WMMA_F32_16X16X4_F32 from an LDS tile (row-major):
// lane<16 : {A[l][4c], A[l][4c+1]}   lane>=16 : {A[l-16][4c+2], A[l-16][4c+3]}
static __device__ inline v2f ldfrag(const float* fbase, int c) {
    const float* p = fbase + 4 * c;
    v2f r; r.x = p[0]; r.y = p[1];
    return r;
}

__global__ __launch_bounds__(128)
void rnn_fused_kernel(const float* __restrict__ x,
                      const float* __restrict__ W_ih0, const float* __restrict__ W_hh0,
                      const float* __restrict__ b_ih0, const float* __restrict__ b_hh0,
                      const float* __restrict__ W_ih1, const float* __restrict__ W_hh1,
                      const float* __restrict__ b_ih1, const float* __restrict__ b_hh1,
                      const float* __restrict__ W_fc,  const float* __restrict__ b_fc,
                      float* __restrict__ out)
{
    __shared__ float lds[LDSFLOATS];

    const int tid  = threadIdx.x;
    const int lane = tid & 31;
    const int wave = tid >> 5;          // 4 waves; wave w owns hidden columns [16w, 16w+16)
    const int cb   = wave * 16;
    const int half = lane >> 4;
    const int l15  = lane & 15;
    const int b0   = blockIdx.x * 16;   // batch-row base for this workgroup

    const int hfb = l15 * HSTR + 2 * half;   // per-lane A-frag base offset (h buffers)
    const int xfb = l15 * XSTR + 2 * half;   // per-lane A-frag base offset (x buffer)

    // ---- persistent weight B-panels in VGPRs: B[k][n] = W[n][k] (D = A*B == h @ W^T) ----
    v2f Bhh0[16], Bih1[16], Bhh1[16], Bih0[3];
    {
        const int col = cb + l15;
        #pragma unroll
        for (int c = 0; c < 16; ++c) {
            const int k = 4 * c + 2 * half;
            Bhh0[c].x = W_hh0[col * HID + k];  Bhh0[c].y = W_hh0[col * HID + k + 1];
            Bih1[c].x = W_ih1[col * HID + k];  Bih1[c].y = W_ih1[col * HID + k + 1];
            Bhh1[c].x = W_hh1[col * HID + k];  Bhh1[c].y = W_hh1[col * HID + k + 1];
        }
        #pragma unroll
        for (int c = 0; c < 3; ++c) {
            const int k = 4 * c + 2 * half;
            Bih0[c].x = (k     < IN) ? W_ih0[col * IN + k]     : 0.0f;
            Bih0[c].y = (k + 1 < IN) ? W_ih0[col * IN + k + 1] : 0.0f;
        }
    }
    const float bias0 = b_ih0[cb + l15] + b_hh0[cb + l15];
    const float bias1 = b_ih1[cb + l15] + b_hh1[cb + l15];

    // Hoisted x-staging offsets: 144 contiguous floats/step -> padded LDS tile.
    const int sj0 = tid;
    const int xo0 = (sj0 / IN) * XSTR + (sj0 % IN);
    const int sj1 = tid + 128;                           // valid only for tid < 16
    const int xo1 = (sj1 / IN) * XSTR + (sj1 % IN);

    // ---- zero h2 ping-pong buffers + x pads; stage x_0 ----
    for (int j = tid; j < 4 * HBUFSZ; j += 128) lds[H1OFF + j] = 0.0f;
    {
        const float* xsrc = x + (size_t)b0 * IN;
        lds[XOFF + xo0] = xsrc[sj0];
        if (tid < 16) lds[XOFF + xo1] = xsrc[sj1];
        for (int j = tid; j < 16 * (XSTR - IN); j += 128) {
            const int r = j / (XSTR - IN), cc = IN + j % (XSTR - IN);
            lds[XOFF + r * XSTR + cc]          = 0.0f;
            lds[XOFF + XBUFSZ + r * XSTR + cc] = 0.0f;
        }
    }
    __syncthreads();

    // h1 A-fragments live in registers across iterations (h1_{t-1}); zero at t=0.
    v2f h1f[16];
    #pragma unroll
    for (int c = 0; c < 16; ++c) { h1f[c].x = 0.0f; h1f[c].y = 0.0f; }

    const float* xgbase = x + ((size_t)BATCH + b0) * IN;  // x[1, b0, 0]

    for (int t = 0; t < SEQ; ++t) {
        const int p = t & 1, q = p ^ 1;
        float* xb  = lds + XOFF  + p * XBUFSZ;
        float* xbn = lds + XOFF  + q * XBUFSZ;
        float* h1p = lds + H1OFF + p * HBUFSZ;
        float* h2p = lds + H2OFF + p * HBUFSZ;
        float* h2q = lds + H2OFF + q * HBUFSZ;

        // -- top of step: issue loads whose data is valid since B2(t-1): --
        //    h2_{t-1} fragments and x_t fragments; consumed >=16 WMMAs later.
        v2f h2f[16], ax[3];
        #pragma unroll
        for (int c = 0; c < 16; ++c) h2f[c] = ldfrag(h2q + hfb, c);
        #pragma unroll
        for (int c = 0; c < 3; ++c)  ax[c]  = ldfrag(xb + xfb, c);

        // stage x_{t+1} (576 contiguous bytes per WG; pads stay zero)
        if (t + 1 < SEQ) {
            const float* xsrc = xgbase + (size_t)t * (BATCH * IN);
            xbn[xo0] = xsrc[sj0];
            if (tid < 16) xbn[xo1] = xsrc[sj1];
        }

        // ---- layer 0: h1_t = tanh(h1_{t-1} @ Whh0^T + x_t @ Wih0^T + bias0) ----
        // hh0 chain consumes only register-resident h1f -> starts with zero waits.
        v8f acc;
        #pragma unroll
        for (int r = 0; r < 8; ++r) acc[r] = bias0;
        #pragma unroll
        for (int c = 0; c < 16; ++c)
            acc = __builtin_amdgcn_wmma_f32_16x16x4_f32(false, h1f[c], false, Bhh0[c],
                                                        (short)0, acc, false, false);
        #pragma unroll
        for (int c = 0; c < 3; ++c)
            acc = __builtin_amdgcn_wmma_f32_16x16x4_f32(false, ax[c], false, Bih0[c],
                                                        (short)0, acc, false, false);
        // pin: all DS reads of this region first, then the 19-WMMA chain
        SCHED_GROUP(0x100, 12, 0);
        SCHED_GROUP(0x008, 19, 0);

        #pragma unroll
        for (int r = 0; r < 8; ++r)
            h1p[(r + 8 * half) * HSTR + cb + l15] = tanh_fast(acc[r]);

        __syncthreads();   // h1_t complete/visible across waves

        // ---- layer 1: h2_t = tanh(h2_{t-1} @ Whh1^T + h1_t @ Wih1^T + bias1) ----
        // Issue fresh h1_t fragment loads (also reused by next step's layer 0),
        // then run the hh1 chain on register-resident h2f to hide their latency.
        #pragma unroll
        for (int c = 0; c < 16; ++c) h1f[c] = ldfrag(h1p + hfb, c);

        #pragma unroll
        for (int r = 0; r < 8; ++r) acc[r] = bias1;
        #pragma unroll
        for (int c = 0; c < 16; ++c)
            acc = __builtin_amdgcn_wmma_f32_16x16x4_f32(false, h2f[c], false, Bhh1[c],
                                                        (short)0, acc, false, false);
        #pragma unroll
        for (int c = 0; c < 16; ++c)
            acc = __builtin_amdgcn_wmma_f32_16x16x4_f32(false, h1f[c], false, Bih1[c],
                                                        (short)0, acc, false, false);
        // pin: 8 ds_load(2addr) first, then 32 WMMAs
        SCHED_GROUP(0x100, 8, 1);
        SCHED_GROUP(0x008, 32, 1);

        #pragma unroll
        for (int r = 0; r < 8; ++r)
            h2p[(r + 8 * half) * HSTR + cb + l15] = tanh_fast(acc[r]);

        __syncthreads();   // h2_t complete/visible; fences next iter's staging/overwrites
    }

    // ---- FC head on final h2 (16 x 10 per WG; trivial VALU work) ----
    const float* hf = lds + H2OFF + ((SEQ - 1) & 1) * HBUFSZ;
    for (int i = tid; i < 16 * OUT; i += 128) {
        const int row = i / OUT, o = i % OUT;
        float s = b_fc[o];
        #pragma unroll 8
        for (int k = 0; k < HID; ++k) s += hf[row * HSTR + k] * W_fc[o * HID + k];
        out[(size_t)(b0 + row) * OUT + o] = s;
    }
}

extern "C" void kernel_launch(void* const* d_in, const int* in_sizes, int n_in,
                              void* d_out, int out_size, void* d_ws, size_t ws_size,
                              hipStream_t stream) {
    const float* x     = (const float*)d_in[0];
    const float* W_ih0 = (const float*)d_in[1];
    const float* W_hh0 = (const float*)d_in[2];
    const float* b_ih0 = (const float*)d_in[3];
    const float* b_hh0 = (const float*)d_in[4];
    const float* W_ih1 = (const float*)d_in[5];
    const float* W_hh1 = (const float*)d_in[6];
    const float* b_ih1 = (const float*)d_in[7];
    const float* b_hh1 = (const float*)d_in[8];
    const float* W_fc  = (const float*)d_in[9];
    const float* b_fc  = (const float*)d_in[10];

    rnn_fused_kernel<<<BATCH / 16, 128, 0, stream>>>(
        x, W_ih0, W_hh0, b_ih0, b_hh0, W_ih1, W_hh1, b_ih1, b_hh1, W_fc, b_fc,
        (float*)d_out);
}